// TurboEncoderDecoder_21723944583245
// MI455X (gfx1250) — compile-verified
//
#include <hip/hip_runtime.h>
#include <stdint.h>

#ifndef __has_builtin
#define __has_builtin(x) 0
#endif

// ---------------------------------------------------------------------------
// Turbo decoder (rate-1/3, RSC (7,5), 4 states, max-log-MAP, 6 iterations)
// One workgroup (1 wave32) per block; everything LDS-resident.
// ---------------------------------------------------------------------------

constexpr int   T     = 2048;   // block length (K)
constexpr int   CHUNK = 64;     // trellis steps per lane (T / 32)
constexpr int   SPAD  = 2176;   // 2048 + 4 floats pad per 64 (keeps 16B alignment)
constexpr float LCF   = 2.0f;   // Lc = 2/sigma^2, sigma = 1
constexpr float NEGF  = -1.0e4f;

// NS[s][u] / PAR[s][u] packed as 2-bit fields, field index = s*2+u
// NS  flat = [0,2,2,0,3,1,1,3] ; PAR flat = [0,1,0,1,1,0,1,0]
constexpr uint32_t NSPACK  = 0xD728u;
constexpr uint32_t PARPACK = 0x1144u;

__device__ __forceinline__ int PADI(int t)  { return t + ((t >> 6) << 2); } // scalar arrays
__device__ __forceinline__ int APADI(int t) { return t + (t >> 6); }        // vector arrays

// ---- CDNA5 async global->LDS staging -------------------------------------
typedef int v4i __attribute__((vector_size(16)));
typedef __attribute__((address_space(1))) v4i* g128_t;
typedef __attribute__((address_space(3))) v4i* l128_t;

__device__ __forceinline__ void ld_async_b128(const void* g, void* l) {
#if __has_builtin(__builtin_amdgcn_global_load_async_to_lds_b128)
  __builtin_amdgcn_global_load_async_to_lds_b128((g128_t)(g), (l128_t)(l), 0, 0);
#else
  asm volatile("global_load_async_to_lds_b128 %0, %1, off"
               :: "v"((uint32_t)(uintptr_t)l), "v"(g) : "memory");
#endif
}

__device__ __forceinline__ void wait_async0() {
#if __has_builtin(__builtin_amdgcn_s_wait_asynccnt)
  __builtin_amdgcn_s_wait_asynccnt(0);
#else
  asm volatile("s_wait_asynccnt 0" ::: "memory");
#endif
}

// ---- 4-state permutation helpers for the RSC encoders --------------------
__device__ __forceinline__ uint32_t perm_step(uint32_t m, int u) {
  uint32_t r = 0;
#pragma unroll
  for (int s = 0; s < 4; ++s) {
    uint32_t os = (m >> (2 * s)) & 3u;
    uint32_t ns = (NSPACK >> (((os << 1) | (uint32_t)u) * 2)) & 3u;
    r |= ns << (2 * s);
  }
  return r;
}
__device__ __forceinline__ uint32_t perm_compose(uint32_t g, uint32_t f) {
  // (g o f)(s) = g[f[s]]  (f applied first)
  uint32_t r = 0;
#pragma unroll
  for (int s = 0; s < 4; ++s) {
    uint32_t fs = (f >> (2 * s)) & 3u;
    r |= ((g >> (2 * fs)) & 3u) << (2 * s);
  }
  return r;
}

// ---- one max-log-MAP BCJR pass over T=2048, chunk-parallel across 32 lanes
// gg[] caches per-step branch metrics (gpp, gpm) so the two replay passes do
// a single ds_load_b64 per step instead of 3 scalar loads + recompute.
__device__ __forceinline__ void bcjr(const float* ys, const float* yp,
                                     const float* la, float* lo,
                                     float4* alp, float2* gg, int lane)
{
  const int t0 = lane * CHUNK;

  // 1) per-lane chunk tropical transition matrix  C[r][c]: state c -> state r
  //    (also writes the gamma cache)
  float C[4][4];
#pragma unroll
  for (int r = 0; r < 4; ++r)
#pragma unroll
    for (int c = 0; c < 4; ++c) C[r][c] = (r == c) ? 0.0f : NEGF;

#pragma unroll 4
  for (int j = 0; j < CHUNK; ++j) {
    int t = t0 + j, p = PADI(t);
    float A   = 0.5f * (LCF * ys[p] + la[p]);
    float Bv  = (0.5f * LCF) * yp[p];
    float gpp = A + Bv;   // u=1, par=1   (g_mm = -gpp)
    float gpm = A - Bv;   // u=1, par=0   (g_mp = -gpm)
    gg[APADI(t)] = make_float2(gpp, gpm);
#pragma unroll
    for (int c = 0; c < 4; ++c) {
      float c0 = C[0][c], c1 = C[1][c], c2 = C[2][c], c3 = C[3][c];
      C[0][c] = fmaxf(c0 - gpp, c1 + gpp);
      C[2][c] = fmaxf(c0 + gpp, c1 - gpp);
      C[1][c] = fmaxf(c2 + gpm, c3 - gpm);
      C[3][c] = fmaxf(c2 - gpm, c3 + gpm);
    }
  }

  float P[4][4], S[4][4];
#pragma unroll
  for (int r = 0; r < 4; ++r)
#pragma unroll
    for (int c = 0; c < 4; ++c) { P[r][c] = C[r][c]; S[r][c] = C[r][c]; }

  // 2) forward Kogge-Stone prefix scan of matrices (tropical compose)
#pragma unroll
  for (int d = 1; d < 32; d <<= 1) {
    float O[4][4];
#pragma unroll
    for (int r = 0; r < 4; ++r)
#pragma unroll
      for (int c = 0; c < 4; ++c) O[r][c] = __shfl_up(P[r][c], d, 32);
    if (lane >= d) {
      float R[4][4];
#pragma unroll
      for (int r = 0; r < 4; ++r)
#pragma unroll
        for (int c = 0; c < 4; ++c)
          R[r][c] = fmaxf(fmaxf(P[r][0] + O[0][c], P[r][1] + O[1][c]),
                          fmaxf(P[r][2] + O[2][c], P[r][3] + O[3][c]));
#pragma unroll
      for (int r = 0; r < 4; ++r)
#pragma unroll
        for (int c = 0; c < 4; ++c) P[r][c] = R[r][c];
    }
  }

  // alpha at chunk start = (exclusive prefix) (x) alpha0, alpha0=(0,NEG,NEG,NEG)
  float a0, a1, a2, a3;
  {
    float Q[4][4];
#pragma unroll
    for (int r = 0; r < 4; ++r)
#pragma unroll
      for (int c = 0; c < 4; ++c) Q[r][c] = __shfl_up(P[r][c], 1, 32);
    if (lane == 0) { a0 = 0.0f; a1 = NEGF; a2 = NEGF; a3 = NEGF; }
    else {
      a0 = fmaxf(fmaxf(Q[0][0], Q[0][1] + NEGF), fmaxf(Q[0][2] + NEGF, Q[0][3] + NEGF));
      a1 = fmaxf(fmaxf(Q[1][0], Q[1][1] + NEGF), fmaxf(Q[1][2] + NEGF, Q[1][3] + NEGF));
      a2 = fmaxf(fmaxf(Q[2][0], Q[2][1] + NEGF), fmaxf(Q[2][2] + NEGF, Q[2][3] + NEGF));
      a3 = fmaxf(fmaxf(Q[3][0], Q[3][1] + NEGF), fmaxf(Q[3][2] + NEGF, Q[3][3] + NEGF));
    }
    float mx = fmaxf(fmaxf(a0, a1), fmaxf(a2, a3));
    a0 -= mx; a1 -= mx; a2 -= mx; a3 -= mx;
  }

  // 3) suffix scan (backward chunk operator = transpose of forward chunk matrix)
#pragma unroll
  for (int d = 1; d < 32; d <<= 1) {
    float O[4][4];
#pragma unroll
    for (int r = 0; r < 4; ++r)
#pragma unroll
      for (int c = 0; c < 4; ++c) O[r][c] = __shfl_down(S[r][c], d, 32);
    if (lane + d < 32) {
      float R[4][4];
#pragma unroll
      for (int r = 0; r < 4; ++r)
#pragma unroll
        for (int c = 0; c < 4; ++c)
          R[r][c] = fmaxf(fmaxf(O[r][0] + S[0][c], O[r][1] + S[1][c]),
                          fmaxf(O[r][2] + S[2][c], O[r][3] + S[3][c]));
#pragma unroll
      for (int r = 0; r < 4; ++r)
#pragma unroll
        for (int c = 0; c < 4; ++c) S[r][c] = R[r][c];
    }
  }

  // beta at chunk end: beta[s] = max_r D[r][s] (beta_T = 0), D = exclusive suffix
  float b0, b1, b2, b3;
  {
    float Q[4][4];
#pragma unroll
    for (int r = 0; r < 4; ++r)
#pragma unroll
      for (int c = 0; c < 4; ++c) Q[r][c] = __shfl_down(S[r][c], 1, 32);
    if (lane == 31) { b0 = b1 = b2 = b3 = 0.0f; }
    else {
      b0 = fmaxf(fmaxf(Q[0][0], Q[1][0]), fmaxf(Q[2][0], Q[3][0]));
      b1 = fmaxf(fmaxf(Q[0][1], Q[1][1]), fmaxf(Q[2][1], Q[3][1]));
      b2 = fmaxf(fmaxf(Q[0][2], Q[1][2]), fmaxf(Q[2][2], Q[3][2]));
      b3 = fmaxf(fmaxf(Q[0][3], Q[1][3]), fmaxf(Q[2][3], Q[3][3]));
    }
    float mx = fmaxf(fmaxf(b0, b1), fmaxf(b2, b3));
    b0 -= mx; b1 -= mx; b2 -= mx; b3 -= mx;
  }

  // 4) forward replay: emit alpha_t (pre-update) into LDS trace
  {
    float r0 = a0, r1 = a1, r2 = a2, r3 = a3;
#pragma unroll 4
    for (int j = 0; j < CHUNK; ++j) {
      int t = t0 + j;
      alp[APADI(t)] = make_float4(r0, r1, r2, r3);
      float2 g = gg[APADI(t)];
      float gpp = g.x, gpm = g.y;
      float f0 = fmaxf(r0 - gpp, r1 + gpp);
      float f2 = fmaxf(r0 + gpp, r1 - gpp);
      float f1 = fmaxf(r2 + gpm, r3 - gpm);
      float f3 = fmaxf(r2 - gpm, r3 + gpm);
      r0 = f0; r1 = f1; r2 = f2; r3 = f3;
    }
  }

  // 5) backward replay + LLR
  {
#pragma unroll 4
    for (int j = CHUNK - 1; j >= 0; --j) {
      int t = t0 + j;
      float2 g = gg[APADI(t)];
      float gpp = g.x, gpm = g.y;
      float4 av = alp[APADI(t)];
      float u1 = gpp + b2, u2 = gpp + b0, u3 = gpm + b1, u4 = gpm + b3; // u=1 edges
      float v1 = b0 - gpp, v2 = b2 - gpp, v3 = b3 - gpm, v4 = b1 - gpm; // u=0 edges
      float m1 = fmaxf(fmaxf(av.x + u1, av.y + u2), fmaxf(av.z + u3, av.w + u4));
      float m0 = fmaxf(fmaxf(av.x + v1, av.y + v2), fmaxf(av.z + v3, av.w + v4));
      lo[PADI(t)] = m1 - m0;
      b0 = fmaxf(v1, u1); b1 = fmaxf(v2, u2);
      b2 = fmaxf(v3, u3); b3 = fmaxf(v4, u4);
    }
  }
}

// ---------------------------------------------------------------------------
__global__ void __launch_bounds__(32)
turbo_decode_kernel(const int* __restrict__ xbits,
                    const float* __restrict__ n1g,
                    const float* __restrict__ n2g,
                    const float* __restrict__ n3g,
                    const int* __restrict__ pig,
                    float* __restrict__ out)
{
  extern __shared__ char smem_raw[];
  float*  Y1  = reinterpret_cast<float*>(smem_raw);
  float*  Y2  = Y1  + SPAD;
  float*  Y1I = Y2  + SPAD;
  float*  Y3  = Y1I + SPAD;
  float*  LA  = Y3  + SPAD;
  float*  LE  = LA  + SPAD;
  int*    PIl = reinterpret_cast<int*>(LE + SPAD);        // T ints
  float4* ALP = reinterpret_cast<float4*>(PIl + T);       // T + 32 float4
  float2* GG  = reinterpret_cast<float2*>(ALP + (T + 32)); // T + 32 float2

  const int    lane = threadIdx.x;
  const size_t base = (size_t)blockIdx.x * T;

  // ---- stage pi, x, n1, n2, n3 into LDS via async b128 (padded layout) ----
#pragma unroll 1
  for (int k = 0; k < 16; ++k) {
    int i4 = (k * 32 + lane) * 4;
    int pp = PADI(i4);
    ld_async_b128(pig + i4,             PIl + i4);
    ld_async_b128(xbits + base + i4,    reinterpret_cast<int*>(LE) + i4);
    ld_async_b128(n1g + base + i4,      Y1 + pp);
    ld_async_b128(n2g + base + i4,      Y2 + pp);
    ld_async_b128(n3g + base + i4,      Y3 + pp);
  }
  wait_async0();
  __syncthreads();

  const int* Xl = reinterpret_cast<const int*>(LE);

  // ---- y1 = bpsk(x) + n1 ----
#pragma unroll 4
  for (int k = 0; k < 64; ++k) {
    int t = k * 32 + lane, p = PADI(t);
    Y1[p] = (Xl[t] ? 1.0f : -1.0f) + Y1[p];
  }

  // ---- RSC encoder 1 (systematic order): y2 = bpsk(par) + n2 ----
  {
    const int t0 = lane * CHUNK;
    uint32_t sg = 0xE4u;                           // identity permutation
#pragma unroll 4
    for (int j = 0; j < CHUNK; ++j) sg = perm_step(sg, Xl[t0 + j] & 1);
#pragma unroll
    for (int d = 1; d < 32; d <<= 1) {
      uint32_t o = __shfl_up(sg, d, 32);
      if (lane >= d) sg = perm_compose(sg, o);
    }
    uint32_t pv = __shfl_up(sg, 1, 32);
    int s = (lane == 0) ? 0 : (int)(pv & 3u);      // state at chunk start (from state 0)
#pragma unroll 4
    for (int j = 0; j < CHUNK; ++j) {
      int t = t0 + j, u = Xl[t] & 1, idx = (s << 1) | u;
      float pb = ((PARPACK >> (idx * 2)) & 1u) ? 1.0f : -1.0f;
      int p = PADI(t);
      Y2[p] = pb + Y2[p];
      s = (int)((NSPACK >> (idx * 2)) & 3u);
    }
  }

  // ---- RSC encoder 2 (interleaved order): y3 = bpsk(par) + n3 ----
  {
    const int t0 = lane * CHUNK;
    uint32_t sg = 0xE4u;
#pragma unroll 4
    for (int j = 0; j < CHUNK; ++j) sg = perm_step(sg, Xl[PIl[t0 + j]] & 1);
#pragma unroll
    for (int d = 1; d < 32; d <<= 1) {
      uint32_t o = __shfl_up(sg, d, 32);
      if (lane >= d) sg = perm_compose(sg, o);
    }
    uint32_t pv = __shfl_up(sg, 1, 32);
    int s = (lane == 0) ? 0 : (int)(pv & 3u);
#pragma unroll 4
    for (int j = 0; j < CHUNK; ++j) {
      int t = t0 + j, u = Xl[PIl[t]] & 1, idx = (s << 1) | u;
      float pb = ((PARPACK >> (idx * 2)) & 1u) ? 1.0f : -1.0f;
      int p = PADI(t);
      Y3[p] = pb + Y3[p];
      s = (int)((NSPACK >> (idx * 2)) & 3u);
    }
  }
  __syncthreads();

  // ---- y1i = gather(y1, pi); La = 0 ----
#pragma unroll 4
  for (int k = 0; k < 64; ++k) {
    int t = k * 32 + lane;
    Y1I[PADI(t)] = Y1[PADI(PIl[t])];
    LA[PADI(t)]  = 0.0f;
  }
  __syncthreads();

  // ---- 6 turbo iterations, fully LDS-resident ----
#pragma unroll 1
  for (int it = 0; it < 6; ++it) {
    bcjr(Y1, Y2, LA, LE, ALP, GG, lane);           // L1 -> LE
    __syncthreads();
#pragma unroll 4
    for (int k = 0; k < 64; ++k) {                 // Le1 = L1 - Lc*y1 - Lint
      int p = PADI(k * 32 + lane);
      LE[p] = LE[p] - LCF * Y1[p] - LA[p];
    }
    __syncthreads();
#pragma unroll 4
    for (int k = 0; k < 64; ++k) {                 // Le1_i = gather(Le1, pi)
      int t = k * 32 + lane;
      LA[PADI(t)] = LE[PADI(PIl[t])];
    }
    __syncthreads();

    bcjr(Y1I, Y3, LA, LE, ALP, GG, lane);          // L2 -> LE
    __syncthreads();
    if (it < 5) {
#pragma unroll 4
      for (int k = 0; k < 64; ++k) {               // Le2 = L2 - Lc*y1i - Le1_i
        int p = PADI(k * 32 + lane);
        LE[p] = LE[p] - LCF * Y1I[p] - LA[p];
      }
      __syncthreads();
#pragma unroll 4
      for (int k = 0; k < 64; ++k) {               // Lint[pi[j]] = Le2[j]
        int t = k * 32 + lane;
        LA[PADI(PIl[t])] = LE[PADI(t)];
      }
      __syncthreads();
    } else {
#pragma unroll 4
      for (int k = 0; k < 64; ++k) {               // out[b, pi[j]] = L2[b, j]
        int t = k * 32 + lane;
        out[base + PIl[t]] = LE[PADI(t)];
      }
    }
  }
}

// ---------------------------------------------------------------------------
static constexpr int SMEM_BYTES =
    6 * SPAD * 4 + T * 4 + (T + 32) * 16 + (T + 32) * 8;   // 110336

extern "C" void kernel_launch(void* const* d_in, const int* in_sizes, int n_in,
                              void* d_out, int out_size, void* d_ws, size_t ws_size,
                              hipStream_t stream) {
  (void)n_in; (void)d_ws; (void)ws_size; (void)out_size;
  const int*   x  = (const int*)d_in[0];
  const float* n1 = (const float*)d_in[1];
  const float* n2 = (const float*)d_in[2];
  const float* n3 = (const float*)d_in[3];
  const int*   pi = (const int*)d_in[4];
  float*       out = (float*)d_out;

  const int K = in_sizes[4];            // 2048
  const int B = in_sizes[0] / K;        // 512

  (void)hipFuncSetAttribute(reinterpret_cast<const void*>(turbo_decode_kernel),
                            hipFuncAttributeMaxDynamicSharedMemorySize, SMEM_BYTES);

  turbo_decode_kernel<<<dim3(B), dim3(32), SMEM_BYTES, stream>>>(x, n1, n2, n3, pi, out);
}